// MTP_9380208575080
// MI455X (gfx1250) — compile-verified
//
#include <hip/hip_runtime.h>

#define BATCH 8
#define NATOM 256
#define MNBR  100
#define NB    25
#define HID   100
#define PI_F  3.141592653589f
#define RMIN_F 0.5f
#define RMAX_F 6.0f

// padded LDS geometry
#define EMB_S    32   // emb row stride; cols 25..31 zero
#define EMB_ROWS 112  // rows 100..111 zero (GEMM2 A M-padding)
#define FCT_ROWS 32   // rows 25..31 zero (GEMM1 A M-padding)
#define DGT_ROWS 28   // rows 25..27 zero (GEMM2 K-padding)

typedef float v2f __attribute__((ext_vector_type(2)));
typedef float v8f __attribute__((ext_vector_type(8)));

__global__ __launch_bounds__(128) void mtp_fused(
    const int* __restrict__ Imagetype,
    const int* __restrict__ neighbor_list,
    const float* __restrict__ ImagedR,
    const float* __restrict__ n_embed,
    const float* __restrict__ W1, const float* __restrict__ b1,
    const float* __restrict__ W2, const float* __restrict__ b2,
    const float* __restrict__ W3, const float* __restrict__ b3,
    const float* __restrict__ W4, const float* __restrict__ b4,
    float* __restrict__ outEtot, float* __restrict__ outEi,
    float* __restrict__ outForce)
{
  __shared__ float fcT [FCT_ROWS * MNBR];   // [k][m]  A of GEMM1
  __shared__ float embL[EMB_ROWS * EMB_S];  // [m][e]  B of GEMM1 / A of GEMM2
  __shared__ float dGT [DGT_ROWS * EMB_S];  // [e][k]  B of GEMM2
  __shared__ float dfcL[MNBR * NB];
  __shared__ float unitL[MNBR * 3];
  __shared__ float GL  [NB * NB];           // 625-vector G
  __shared__ float dEL [MNBR * NB];
  __shared__ float x1L[HID], t1L[HID], x3L[HID], t3L[HID], t5L[HID];
  __shared__ float g5L[HID], g3L[HID], g1L[HID];
  __shared__ float fredL[MNBR * 3];

  const int tid  = threadIdx.x;
  const int lane = tid & 31;
  const int wave = tid >> 5;
  const int l16  = lane & 15;
  const int lhi  = lane >> 4;      // 0 or 1
  const int lh2  = lhi * 2;        // 0 or 2

  const int bi = blockIdx.x / NATOM;
  const int ni = blockIdx.x % NATOM;
  const size_t atom = (size_t)bi * NATOM + ni;

  // zero padded tiles
  for (int i = tid; i < FCT_ROWS * MNBR; i += 128) fcT[i]  = 0.f;
  for (int i = tid; i < EMB_ROWS * EMB_S; i += 128) embL[i] = 0.f;
  for (int i = tid; i < DGT_ROWS * EMB_S; i += 128) dGT[i]  = 0.f;
  __syncthreads();

  // ---------- phase 1: fc, dfc, unit, embedding outer product ----------
  const int it = Imagetype[atom];
  float iit[5];
  #pragma unroll
  for (int i = 0; i < 5; ++i) iit[i] = n_embed[it * 5 + i];

  for (int m = tid; m < MNBR; m += 128) {
    const float* p = ImagedR + (atom * MNBR + m) * 4;
    float r = p[0], x = p[1], y = p[2], z = p[3];
    int nbr = neighbor_list[atom * MNBR + m];
    int jt  = (nbr == 0) ? 0 : Imagetype[(size_t)bi * NATOM + (nbr - 1)];
    #pragma unroll
    for (int i = 0; i < 5; ++i) {
      float a = iit[i];
      #pragma unroll
      for (int j = 0; j < 5; ++j)
        embL[m * EMB_S + i * 5 + j] = a * n_embed[jt * 5 + j];
    }
    bool  cond = (r < RMAX_F) && (r > 0.f);
    float cutv = 0.5f * cosf(PI_F * r) + 0.5f;
    float sp   = 0.5f * PI_F * sinf(PI_F * r);
    float rinv = (r != 0.f) ? 1.f / r : 0.f;
    unitL[m * 3 + 0] = x * rinv;
    unitL[m * 3 + 1] = y * rinv;
    unitL[m * 3 + 2] = z * rinv;
    #pragma unroll
    for (int k = 0; k < NB; ++k) {
      float rs = RMIN_F + (float)k * ((RMAX_F - RMIN_F) / (float)(NB - 1));
      float d = r - rs;
      float g = expf(-d * d);
      fcT[k * MNBR + m] = cond ? g * cutv : 0.f;
      dfcL[m * NB + k]  = cond ? (g * (-2.f * d) * cutv - g * sp) : 0.f;
    }
  }
  __syncthreads();

  // ---------- phase 2: G[k][e] = sum_m fcT[k][m] * emb[m][e]  (WMMA f32) ----
  {
    const int Mt = wave >> 1, Nt = wave & 1;        // 4 waves -> 4 tiles
    v8f acc = {0.f,0.f,0.f,0.f,0.f,0.f,0.f,0.f};
    const int arow = Mt * 16 + l16;
    const int bcol = Nt * 16 + l16;
    for (int kk = 0; kk < MNBR; kk += 4) {
      v2f a, bb;
      a.x  = fcT[arow * MNBR + kk + lh2];
      a.y  = fcT[arow * MNBR + kk + lh2 + 1];
      bb.x = embL[(kk + lh2)     * EMB_S + bcol];
      bb.y = embL[(kk + lh2 + 1) * EMB_S + bcol];
      acc = __builtin_amdgcn_wmma_f32_16x16x4_f32(
              false, a, false, bb, (short)0, acc, false, false);
    }
    #pragma unroll
    for (int v = 0; v < 8; ++v) {
      int row = Mt * 16 + v + lhi * 8;
      if (row < NB && bcol < NB) GL[row * NB + bcol] = acc[v];
    }
  }
  __syncthreads();

  // ---------- phase 3: MLP forward ----------
  if (tid < HID) {
    float s = b1[tid];
    const float* w = W1 + (size_t)tid * 625;
    for (int i = 0; i < 625; ++i) s += w[i] * GL[i];
    x1L[tid] = s; t1L[tid] = tanhf(s);
  }
  __syncthreads();
  if (tid < HID) {
    float s = b2[tid] + x1L[tid];
    const float* w = W2 + tid * HID;
    for (int i = 0; i < HID; ++i) s += w[i] * t1L[i];
    x3L[tid] = s; t3L[tid] = tanhf(s);
  }
  __syncthreads();
  if (tid < HID) {
    float s = b3[tid] + x3L[tid];
    const float* w = W3 + tid * HID;
    for (int i = 0; i < HID; ++i) s += w[i] * t3L[i];
    t5L[tid] = tanhf(s);
  }
  __syncthreads();
  if (tid == 0) {
    float e = b4[0];
    for (int j = 0; j < HID; ++j) e += W4[j] * t5L[j];
    outEi[atom] = e;
    atomicAdd(&outEtot[bi], e);
  }

  // ---------- phase 4: MLP backward ----------
  if (tid < HID) { float t = t5L[tid]; g5L[tid] = W4[tid] * (1.f - t * t); }
  __syncthreads();
  if (tid < HID) {
    float s = 0.f;
    for (int i = 0; i < HID; ++i) s += g5L[i] * W3[i * HID + tid];
    float t = t3L[tid];
    g3L[tid] = g5L[tid] + s * (1.f - t * t);
  }
  __syncthreads();
  if (tid < HID) {
    float s = 0.f;
    for (int i = 0; i < HID; ++i) s += g3L[i] * W2[i * HID + tid];
    float t = t1L[tid];
    g1L[tid] = g3L[tid] + s * (1.f - t * t);
  }
  __syncthreads();
  for (int i = tid; i < 625; i += 128) {        // dG = g1 @ W1, stored transposed
    float s = 0.f;
    for (int j = 0; j < HID; ++j) s += g1L[j] * W1[(size_t)j * 625 + i];
    dGT[(i % NB) * EMB_S + (i / NB)] = s;
  }
  __syncthreads();

  // ---------- phase 5: dE[m][k] = sum_e emb[m][e] * dGT[e][k]  (WMMA f32) ----
  for (int t = wave; t < 14; t += 4) {           // 7 Mtiles x 2 Ntiles
    const int Mt = t >> 1, Nt = t & 1;
    v8f acc = {0.f,0.f,0.f,0.f,0.f,0.f,0.f,0.f};
    const int arow = Mt * 16 + l16;
    const int bcol = Nt * 16 + l16;
    for (int kk = 0; kk < DGT_ROWS; kk += 4) {
      v2f a, bb;
      a.x  = embL[arow * EMB_S + kk + lh2];
      a.y  = embL[arow * EMB_S + kk + lh2 + 1];
      bb.x = dGT[(kk + lh2)     * EMB_S + bcol];
      bb.y = dGT[(kk + lh2 + 1) * EMB_S + bcol];
      acc = __builtin_amdgcn_wmma_f32_16x16x4_f32(
              false, a, false, bb, (short)0, acc, false, false);
    }
    #pragma unroll
    for (int v = 0; v < 8; ++v) {
      int row = Mt * 16 + v + lhi * 8;
      if (row < MNBR && bcol < NB) dEL[row * NB + bcol] = acc[v];
    }
  }
  __syncthreads();

  // ---------- phase 6: forces ----------
  for (int m = tid; m < MNBR; m += 128) {
    float s = 0.f;
    for (int k = 0; k < NB; ++k) s += dEL[m * NB + k] * dfcL[m * NB + k];
    float fx = s * unitL[m * 3 + 0];
    float fy = s * unitL[m * 3 + 1];
    float fz = s * unitL[m * 3 + 2];
    fredL[m * 3 + 0] = fx; fredL[m * 3 + 1] = fy; fredL[m * 3 + 2] = fz;
    int nbr = neighbor_list[atom * MNBR + m];
    if (nbr >= 1) {
      float* fp = outForce + ((size_t)bi * NATOM + nbr - 1) * 3;
      atomicAdd(fp + 0, fx);
      atomicAdd(fp + 1, fy);
      atomicAdd(fp + 2, fz);
    }
  }
  __syncthreads();
  if (tid < 3) {
    float s = 0.f;
    for (int m = 0; m < MNBR; ++m) s += fredL[m * 3 + tid];
    atomicAdd(&outForce[atom * 3 + tid], -s);
  }
}

__global__ void zero_out(float* __restrict__ out, int n) {
  for (int i = blockIdx.x * blockDim.x + threadIdx.x; i < n;
       i += gridDim.x * blockDim.x)
    out[i] = 0.f;
}

extern "C" void kernel_launch(void* const* d_in, const int* in_sizes, int n_in,
                              void* d_out, int out_size, void* d_ws, size_t ws_size,
                              hipStream_t stream) {
  (void)in_sizes; (void)n_in; (void)d_ws; (void)ws_size; (void)out_size;
  float* out = (float*)d_out;
  const int total = BATCH + BATCH * NATOM + BATCH * NATOM * 3; // Etot|Ei|Force
  zero_out<<<32, 256, 0, stream>>>(out, total);
  mtp_fused<<<BATCH * NATOM, 128, 0, stream>>>(
      (const int*)d_in[0], (const int*)d_in[1],
      (const float*)d_in[2], (const float*)d_in[3],
      (const float*)d_in[4], (const float*)d_in[5],
      (const float*)d_in[6], (const float*)d_in[7],
      (const float*)d_in[8], (const float*)d_in[9],
      (const float*)d_in[10], (const float*)d_in[11],
      out,                      // Etot (8)
      out + BATCH,              // Ei   (2048)
      out + BATCH + BATCH * NATOM); // Force (6144)
}